// GCN1_35545149342390
// MI455X (gfx1250) — compile-verified
//
#include <hip/hip_runtime.h>
#include <hip/hip_bf16.h>
#include <math.h>

typedef __attribute__((ext_vector_type(2))) float v2f;
typedef __attribute__((ext_vector_type(8))) float v8f;

#define N_NODES 100000
#define N_EDGES 800000
#define F_IN    128
#define F_HID   256
#define N_CLS   40

// ---------------------------------------------------------------------------
// fp32 WMMA GEMM: C[M,N] = A[M,K] * B[K,N] (+bias, +relu)
// B is pre-transposed + zero-padded into Bt[Npad][K] (Npad multiple of 64):
// guard-free full-EXEC inner loop, one aligned global_load_b64 per fragment.
// Software-pipelined one k-step ahead so fragment loads are a full iteration
// older than their consuming WMMA (hides L2 latency instead of waiting to 0).
// Fragment layouts per CDNA5 ISA 7.12.2 (V_WMMA_F32_16X16X4_F32):
//   A 16x4 : lanes 0-15 hold M=lane,K={0,1}; lanes 16-31 hold M=lane-16,K={2,3}
//   B 4x16 : lanes 0-15 hold N=lane,K={0,1}; lanes 16-31 hold N=lane-16,K={2,3}
//   C 16x16: VGPR i -> row (i + 8*half), col = lane&15
// ---------------------------------------------------------------------------
template<bool RELU, bool HAS_BIAS, int K>
__global__ __launch_bounds__(128) void gemm_wmma_f32(
    const float* __restrict__ A, const float* __restrict__ Bt,
    const float* __restrict__ bias, float* __restrict__ C,
    int M, int N)
{
    const int lane = threadIdx.x & 31;
    const int wave = threadIdx.x >> 5;
    const int half = lane >> 4;    // 0 => K low half, 1 => K high half
    const int l16  = lane & 15;

    const int m0 = (blockIdx.x * 4 + wave) * 16;
    if (m0 >= M) return;                   // wave-uniform exit

    const int n0 = blockIdx.y * 64;

    v8f acc0 = {}, acc1 = {}, acc2 = {}, acc3 = {};

    const float* ap  = A  + (size_t)(m0 + l16) * K + 2 * half;
    const float* bp0 = Bt + (size_t)(n0 +  0 + l16) * K + 2 * half;
    const float* bp1 = Bt + (size_t)(n0 + 16 + l16) * K + 2 * half;
    const float* bp2 = Bt + (size_t)(n0 + 32 + l16) * K + 2 * half;
    const float* bp3 = Bt + (size_t)(n0 + 48 + l16) * K + 2 * half;

    // prologue: load first fragments
    v2f a  = *(const v2f*)(ap);
    v2f b0 = *(const v2f*)(bp0);
    v2f b1 = *(const v2f*)(bp1);
    v2f b2 = *(const v2f*)(bp2);
    v2f b3 = *(const v2f*)(bp3);

#pragma unroll
    for (int k = 0; k < K - 4; k += 4) {
        // issue next step's loads before consuming current fragments
        v2f an  = *(const v2f*)(ap  + k + 4);
        v2f bn0 = *(const v2f*)(bp0 + k + 4);
        v2f bn1 = *(const v2f*)(bp1 + k + 4);
        v2f bn2 = *(const v2f*)(bp2 + k + 4);
        v2f bn3 = *(const v2f*)(bp3 + k + 4);

        acc0 = __builtin_amdgcn_wmma_f32_16x16x4_f32(false, a, false, b0, (short)0, acc0, false, false);
        acc1 = __builtin_amdgcn_wmma_f32_16x16x4_f32(false, a, false, b1, (short)0, acc1, false, false);
        acc2 = __builtin_amdgcn_wmma_f32_16x16x4_f32(false, a, false, b2, (short)0, acc2, false, false);
        acc3 = __builtin_amdgcn_wmma_f32_16x16x4_f32(false, a, false, b3, (short)0, acc3, false, false);

        a = an; b0 = bn0; b1 = bn1; b2 = bn2; b3 = bn3;
    }
    // epilogue step
    acc0 = __builtin_amdgcn_wmma_f32_16x16x4_f32(false, a, false, b0, (short)0, acc0, false, false);
    acc1 = __builtin_amdgcn_wmma_f32_16x16x4_f32(false, a, false, b1, (short)0, acc1, false, false);
    acc2 = __builtin_amdgcn_wmma_f32_16x16x4_f32(false, a, false, b2, (short)0, acc2, false, false);
    acc3 = __builtin_amdgcn_wmma_f32_16x16x4_f32(false, a, false, b3, (short)0, acc3, false, false);

    v8f accs[4] = {acc0, acc1, acc2, acc3};
#pragma unroll
    for (int t = 0; t < 4; ++t) {
        const int n = n0 + t * 16 + l16;
        if (n < N) {
            const float bv = HAS_BIAS ? bias[n] : 0.0f;
#pragma unroll
            for (int i = 0; i < 8; ++i) {
                float v = accs[t][i] + bv;
                if (RELU) v = fmaxf(v, 0.0f);
                C[(size_t)(m0 + i + 8 * half) * N + n] = v;
            }
        }
    }
}

// Bt[n][k] = (n < N) ? B[k][n] : 0   over Npad*K elements
__global__ __launch_bounds__(256) void transpose_pad_kernel(const float* __restrict__ B,
                                                            float* __restrict__ Bt,
                                                            int K, int N, int Npad)
{
    int idx = blockIdx.x * blockDim.x + threadIdx.x;
    if (idx < Npad * K) {
        int n = idx / K;
        int k = idx - n * K;
        Bt[idx] = (n < N) ? B[(size_t)k * N + n] : 0.0f;
    }
}

// in-degree from edges (self-loop added analytically later)
__global__ __launch_bounds__(256) void deg_kernel(const long long* __restrict__ ei,
                                                  float* __restrict__ deg, int nE)
{
    int e = blockIdx.x * blockDim.x + threadIdx.x;
    if (e < nE) atomicAdd(&deg[(int)ei[(size_t)nE + e]], 1.0f);
}

// dinv = rsqrt(deg_edges + 1)
__global__ __launch_bounds__(256) void dinv_kernel(const float* __restrict__ deg,
                                                   float* __restrict__ dinv, int n)
{
    int i = blockIdx.x * blockDim.x + threadIdx.x;
    if (i < n) dinv[i] = rsqrtf(deg[i] + 1.0f);
}

// agg[i,:] = h[i,:] * dinv[i]^2   (self-loop message; fully initializes agg)
__global__ __launch_bounds__(256) void selfloop_init(const float* __restrict__ h,
                                                     const float* __restrict__ dinv,
                                                     float* __restrict__ agg)
{
    size_t idx = (size_t)blockIdx.x * blockDim.x + threadIdx.x;
    if (idx < (size_t)N_NODES * F_HID) {
        int i = (int)(idx >> 8);           // / F_HID (=256)
        float di = dinv[i];
        agg[idx] = h[idx] * di * di;
    }
}

// one block (256 threads = F_HID) per edge: agg[dst,:] += h[src,:] * norm
__global__ __launch_bounds__(256) void scatter_kernel(const long long* __restrict__ ei,
                                                      const float* __restrict__ h,
                                                      const float* __restrict__ dinv,
                                                      float* __restrict__ agg, int nE)
{
    const int e = blockIdx.x;
    const int s = (int)ei[e];
    const int d = (int)ei[(size_t)nE + e];
    const float norm = dinv[s] * dinv[d];
    const int f = threadIdx.x;
    atomicAdd(&agg[(size_t)d * F_HID + f], h[(size_t)s * F_HID + f] * norm);
}

// h1 = relu(agg + b)
__global__ __launch_bounds__(256) void bias_relu_kernel(const float* __restrict__ agg,
                                                        const float* __restrict__ b,
                                                        float* __restrict__ out)
{
    size_t idx = (size_t)blockIdx.x * blockDim.x + threadIdx.x;
    if (idx < (size_t)N_NODES * F_HID) {
        out[idx] = fmaxf(agg[idx] + b[idx & (F_HID - 1)], 0.0f);
    }
}

// in-place row-wise log_softmax over N_CLS columns
__global__ __launch_bounds__(256) void logsoftmax_kernel(float* __restrict__ x, int n)
{
    int i = blockIdx.x * blockDim.x + threadIdx.x;
    if (i >= n) return;
    float* row = x + (size_t)i * N_CLS;
    float m = row[0];
#pragma unroll
    for (int c = 1; c < N_CLS; ++c) m = fmaxf(m, row[c]);
    float s = 0.0f;
#pragma unroll
    for (int c = 0; c < N_CLS; ++c) s += __expf(row[c] - m);
    float ls = __logf(s);
#pragma unroll
    for (int c = 0; c < N_CLS; ++c) row[c] = row[c] - m - ls;
}

extern "C" void kernel_launch(void* const* d_in, const int* in_sizes, int n_in,
                              void* d_out, int out_size, void* d_ws, size_t ws_size,
                              hipStream_t stream)
{
    const float*     x  = (const float*)d_in[0];
    const long long* ei = (const long long*)d_in[1];   // int64 [2, N_EDGES]
    const float*     W  = (const float*)d_in[2];
    const float*     b  = (const float*)d_in[3];
    const float*     W1 = (const float*)d_in[4];
    const float*     b1 = (const float*)d_in[5];
    const float*     W2 = (const float*)d_in[6];
    const float*     b2 = (const float*)d_in[7];
    const float*     W3 = (const float*)d_in[8];
    const float*     b3 = (const float*)d_in[9];
    float* out = (float*)d_out;

    char* ws = (char*)d_ws;
    const size_t hBytes = (size_t)N_NODES * F_HID * sizeof(float);   // 102.4 MB
    float* h    = (float*)(ws);                       // [N, 256] ; later reused as h3 [N,64]
    float* agg  = (float*)(ws + hBytes);              // [N, 256] ; later reused as h2 [N,128]
    size_t off  = 2 * hBytes;
    float* deg  = (float*)(ws + off);  off += (size_t)N_NODES * sizeof(float);
    float* dinv = (float*)(ws + off);  off += (size_t)N_NODES * sizeof(float);
    float* Bt0  = (float*)(ws + off);  off += (size_t)256 * F_IN  * sizeof(float); // W  : [256][128]
    float* Bt1  = (float*)(ws + off);  off += (size_t)128 * F_HID * sizeof(float); // W1 : [128][256]
    float* Bt2  = (float*)(ws + off);  off += (size_t)64  * 128   * sizeof(float); // W2 : [64][128]
    float* Bt3  = (float*)(ws + off);  off += (size_t)64  * 64    * sizeof(float); // W3p: [64][64]
    float* h2 = agg;
    float* h3 = h;

    const int mblocks = (N_NODES / 16 + 3) / 4;       // 1563 (100000 = 16*6250)

    // pack weights: transposed + zero-padded
    transpose_pad_kernel<<<(256 * F_IN  + 255) / 256, 256, 0, stream>>>(W,  Bt0, F_IN,  F_HID, 256);
    transpose_pad_kernel<<<(128 * F_HID + 255) / 256, 256, 0, stream>>>(W1, Bt1, F_HID, 128,   128);
    transpose_pad_kernel<<<(64  * 128   + 255) / 256, 256, 0, stream>>>(W2, Bt2, 128,   64,    64);
    transpose_pad_kernel<<<(64  * 64    + 255) / 256, 256, 0, stream>>>(W3, Bt3, 64,    N_CLS, 64);

    // degree / normalization
    hipMemsetAsync(deg, 0, (size_t)N_NODES * sizeof(float), stream);
    deg_kernel<<<(N_EDGES + 255) / 256, 256, 0, stream>>>(ei, deg, N_EDGES);
    dinv_kernel<<<(N_NODES + 255) / 256, 256, 0, stream>>>(deg, dinv, N_NODES);

    // h = x @ W                                  [1e5,128]x[128,256]
    gemm_wmma_f32<false, false, F_IN><<<dim3(mblocks, 4), 128, 0, stream>>>(
        x, Bt0, nullptr, h, N_NODES, F_HID);

    // agg = self-loop term, then scatter-add edge messages
    selfloop_init<<<(unsigned)(((size_t)N_NODES * F_HID + 255) / 256), 256, 0, stream>>>(h, dinv, agg);
    scatter_kernel<<<N_EDGES, F_HID, 0, stream>>>(ei, h, dinv, agg, N_EDGES);

    // h1 = relu(agg + b)  (into h buffer)
    bias_relu_kernel<<<(unsigned)(((size_t)N_NODES * F_HID + 255) / 256), 256, 0, stream>>>(agg, b, h);

    // h2 = relu(h1 @ W1 + b1)                    [1e5,256]x[256,128]
    gemm_wmma_f32<true, true, F_HID><<<dim3(mblocks, 2), 128, 0, stream>>>(
        h, Bt1, b1, h2, N_NODES, 128);

    // h3 = relu(h2 @ W2 + b2)                    [1e5,128]x[128,64]
    gemm_wmma_f32<true, true, 128><<<dim3(mblocks, 1), 128, 0, stream>>>(
        h2, Bt2, b2, h3, N_NODES, 64);

    // logits = h3 @ W3 + b3 -> d_out             [1e5,64]x[64,40]
    gemm_wmma_f32<false, true, 64><<<dim3(mblocks, 1), 128, 0, stream>>>(
        h3, Bt3, b3, out, N_NODES, N_CLS);

    // log_softmax in place
    logsoftmax_kernel<<<(N_NODES + 255) / 256, 256, 0, stream>>>(out, N_NODES);
}